// UNet_72430328480154
// MI455X (gfx1250) — compile-verified
//
#include <hip/hip_runtime.h>

#define HH 1024
#define WW 1024
#define TILE 32

typedef __attribute__((ext_vector_type(2))) float v2f;
typedef __attribute__((ext_vector_type(8))) float v8f;

__device__ __forceinline__ float sigmoidf_(float v) {
    return 1.0f / (1.0f + __expf(-v));
}

// Fold the two linear layers (no nonlinearity between them in the reference):
//   outfeature = diff @ Mt + c   with Mt = lw1^T @ lw2^T  (6x2), c = lw2@lb1 + lb2
// Stored zero-padded to 8(K) x 16(N) row-major in ws[0..127], biases ws[128..129].
__global__ void prep_fold(const float* __restrict__ lw1, const float* __restrict__ lb1,
                          const float* __restrict__ lw2, const float* __restrict__ lb2,
                          float* __restrict__ ws) {
    int t = threadIdx.x;
    if (t < 128) {
        int k = t >> 4, n = t & 15;
        float acc = 0.0f;
        if (k < 6 && n < 2) {
            for (int h = 0; h < 128; ++h)
                acc += lw1[h * 6 + k] * lw2[n * 128 + h];
        }
        ws[t] = acc;
    } else if (t < 130) {
        int n = t - 128;
        float acc = lb2[n];
        for (int h = 0; h < 128; ++h)
            acc += lw2[n * 128 + h] * lb1[h];
        ws[128 + n] = acc;
    }
}

// ---- shared-memory layout (flat, overlaid where lifetimes allow) ----
//  xs  : 3*38*39 = 4446 floats at 0        (input tile + halo 3)
//  c4s : 36*37   = 1332 floats at 4446
//  c5s : 34*35   = 1190 floats at 5778
//  den : 1024*8  = 8192 floats at 6968     (8ch: 6 data + 2 zero pad for WMMA K)
//  outs: 1024*2  = 2048 floats at 4446     (overlays dead c4s/c5s after convs)
//  total 15160 floats = 60640 B  (< 64 KB static LDS)
#define SMEM_FLOATS 15160
#define XS(ch,r,c)  smem[(ch)*(38*39) + (r)*39 + (c)]
#define C4(r,c)     smem[4446 + (r)*37 + (c)]
#define C5(r,c)     smem[5778 + (r)*35 + (c)]
#define DEN(p,k)    smem[6968 + ((p) << 3) + (k)]
#define OUTS(p,k)   smem[4446 + ((p) << 1) + (k)]

__global__ __launch_bounds__(256)
void encode_mlp(const float* __restrict__ xA, const float* __restrict__ xB,
                const float* __restrict__ w1a, const float* __restrict__ b1a,
                const float* __restrict__ w2a, const float* __restrict__ b2a,
                const float* __restrict__ w3a, const float* __restrict__ b3a,
                const float* __restrict__ w1b, const float* __restrict__ b1b,
                const float* __restrict__ w2b, const float* __restrict__ b2b,
                const float* __restrict__ w3b, const float* __restrict__ b3b,
                const float* __restrict__ fold,   // ws: 8x16 Mt + 2 biases
                float* __restrict__ outf,         // [HW,2]
                float* __restrict__ den1g,        // [H,W,6]
                float* __restrict__ den2g)        // [H,W,6]
{
    __shared__ float smem[SMEM_FLOATS];

    const int tid = threadIdx.x;
    const int X = blockIdx.x * TILE;
    const int Y = blockIdx.y * TILE;

    #pragma unroll 1
    for (int br = 0; br < 2; ++br) {
        const float* xp = br ? xB : xA;
        const float* w1 = br ? w1b : w1a;  const float b1 = br ? b1b[0] : b1a[0];
        const float* w2 = br ? w2b : w2a;  const float b2 = br ? b2b[0] : b2a[0];
        const float* w3 = br ? w3b : w3a;  const float b3 = br ? b3b[0] : b3a[0];
        float* deng = br ? den2g : den1g;

        __syncthreads();  // WAR guard before overwriting xs/c4s/c5s

        // ---- load input tile (halo 3), SAME zero padding ----
        for (int i = tid; i < 3 * 38 * 38; i += 256) {
            int ch = i / (38 * 38);
            int rem = i - ch * (38 * 38);
            int r = rem / 38, c = rem - r * 38;
            int gy = Y - 3 + r, gx = X - 3 + c;
            float v = 0.0f;
            if ((unsigned)gy < (unsigned)HH && (unsigned)gx < (unsigned)WW)
                v = xp[ch * (HH * WW) + gy * WW + gx];
            XS(ch, r, c) = v;
        }
        // prefetch branch-b tile into caches while branch-a computes
        if (br == 0) {
            for (int i = tid; i < 3 * 38; i += 256) {
                int ch = i / 38, r = i - ch * 38;
                int gy = Y - 3 + r;
                if ((unsigned)gy < (unsigned)HH) {
                    int gx = X - 3; if (gx < 0) gx = 0;
                    const float* p = xB + ch * (HH * WW) + gy * WW + gx;
                    __builtin_prefetch(p, 0, 0);
                    __builtin_prefetch(p + 32, 0, 0);   // cover 152 B row span
                }
            }
        }
        __syncthreads();

        // ---- stage 1: c4 = sigmoid(conv1(x)) over 36x36 ----
        {
            float Wk[27];
            #pragma unroll
            for (int j = 0; j < 27; ++j) Wk[j] = w1[j];
            for (int i = tid; i < 36 * 36; i += 256) {
                int r = i / 36, c = i - r * 36;
                float acc = b1;
                #pragma unroll
                for (int ch = 0; ch < 3; ++ch)
                    #pragma unroll
                    for (int ky = 0; ky < 3; ++ky)
                        #pragma unroll
                        for (int kx = 0; kx < 3; ++kx)
                            acc += Wk[ch * 9 + ky * 3 + kx] * XS(ch, r + ky, c + kx);
                C4(r, c) = sigmoidf_(acc);
            }
        }
        __syncthreads();

        // ---- stage 2: c5 = sigmoid(conv2([x, c4])) over 34x34 ----
        {
            float Wk[36];
            #pragma unroll
            for (int j = 0; j < 36; ++j) Wk[j] = w2[j];
            for (int i = tid; i < 34 * 34; i += 256) {
                int r = i / 34, c = i - r * 34;
                float acc = b2;
                #pragma unroll
                for (int ch = 0; ch < 3; ++ch)
                    #pragma unroll
                    for (int ky = 0; ky < 3; ++ky)
                        #pragma unroll
                        for (int kx = 0; kx < 3; ++kx)
                            acc += Wk[ch * 9 + ky * 3 + kx] * XS(ch, r + 1 + ky, c + 1 + kx);
                #pragma unroll
                for (int ky = 0; ky < 3; ++ky)
                    #pragma unroll
                    for (int kx = 0; kx < 3; ++kx)
                        acc += Wk[27 + ky * 3 + kx] * C4(r + ky, c + kx);
                C5(r, c) = sigmoidf_(acc);
            }
        }
        __syncthreads();

        // ---- stage 3: c6 = sigmoid(conv3([x, c4, c5])) over 32x32; emit den ----
        {
            float Wk[45];
            #pragma unroll
            for (int j = 0; j < 45; ++j) Wk[j] = w3[j];
            for (int i = tid; i < TILE * TILE; i += 256) {
                int r = i >> 5, c = i & 31;
                float acc = b3;
                #pragma unroll
                for (int ch = 0; ch < 3; ++ch)
                    #pragma unroll
                    for (int ky = 0; ky < 3; ++ky)
                        #pragma unroll
                        for (int kx = 0; kx < 3; ++kx)
                            acc += Wk[ch * 9 + ky * 3 + kx] * XS(ch, r + 2 + ky, c + 2 + kx);
                #pragma unroll
                for (int ky = 0; ky < 3; ++ky)
                    #pragma unroll
                    for (int kx = 0; kx < 3; ++kx)
                        acc += Wk[27 + ky * 3 + kx] * C4(r + 1 + ky, c + 1 + kx);
                #pragma unroll
                for (int ky = 0; ky < 3; ++ky)
                    #pragma unroll
                    for (int kx = 0; kx < 3; ++kx)
                        acc += Wk[36 + ky * 3 + kx] * C5(r + ky, c + kx);

                float v5 = sigmoidf_(acc);
                float v0 = XS(0, r + 3, c + 3);
                float v1 = XS(1, r + 3, c + 3);
                float v2 = XS(2, r + 3, c + 3);
                float v3 = C4(r + 2, c + 2);
                float v4 = C5(r + 1, c + 1);

                long gpix = (long)(Y + r) * WW + (X + c);
                float* dst = deng + gpix * 6;          // 8B aligned (24B stride)
                *(float2*)(dst + 0) = make_float2(v0, v1);
                *(float2*)(dst + 2) = make_float2(v2, v3);
                *(float2*)(dst + 4) = make_float2(v4, v5);

                if (br == 0) {
                    DEN(i, 0) = v0; DEN(i, 1) = v1; DEN(i, 2) = v2;
                    DEN(i, 3) = v3; DEN(i, 4) = v4; DEN(i, 5) = v5;
                    DEN(i, 6) = 0.0f; DEN(i, 7) = 0.0f;   // WMMA K padding
                } else {
                    DEN(i, 0) -= v0; DEN(i, 1) -= v1; DEN(i, 2) -= v2;
                    DEN(i, 3) -= v3; DEN(i, 4) -= v4; DEN(i, 5) -= v5;
                }
            }
        }
    }
    __syncthreads();

    // ---- folded 6->2 linear on diff via V_WMMA_F32_16X16X4_F32 ----
    // f32 A 16x4 layout: lanes 0-15 hold M rows, VGPR0={K0|K2}, VGPR1={K1|K3}
    // (low|high half-wave). K padded 6->8 => two chained WMMAs. Channels 6,7 of
    // den are pre-zeroed so both A operands are plain aligned ds_load_b64.
    const int lane = tid & 31;
    const int wv   = tid >> 5;
    const bool hi  = lane >= 16;
    const int nn   = lane & 15;

    v2f B0, B1;
    B0.x = fold[(hi ? 2 : 0) * 16 + nn];
    B0.y = fold[(hi ? 3 : 1) * 16 + nn];
    B1.x = fold[(hi ? 6 : 4) * 16 + nn];
    B1.y = fold[(hi ? 7 : 5) * 16 + nn];
    const float bias = (nn == 0) ? fold[128] : ((nn == 1) ? fold[129] : 0.0f);
    v8f Cb = {bias, bias, bias, bias, bias, bias, bias, bias};

    #pragma unroll 1
    for (int b = 0; b < 8; ++b) {              // 8 waves x 8 blocks x 16 px = 1024 px
        const int p0  = wv * 128 + b * 16;
        const int pix = p0 + nn;
        v2f A0 = *(const v2f*)&DEN(pix, hi ? 2 : 0);   // K0,K1 | K2,K3
        v2f A1 = *(const v2f*)&DEN(pix, hi ? 6 : 4);   // K4,K5 | 0,0

        v8f D = __builtin_amdgcn_wmma_f32_16x16x4_f32(
            false, A1, false, B1, (short)0, Cb, false, false);
        D = __builtin_amdgcn_wmma_f32_16x16x4_f32(
            false, A0, false, B0, (short)0, D, false, false);

        if (nn < 2) {                           // only N=0,1 are real outputs
            #pragma unroll
            for (int j = 0; j < 8; ++j) {
                int pit = p0 + j + (hi ? 8 : 0);   // D: VGPR j => row j (low) / j+8 (high)
                OUTS(pit, nn) = D[j];
            }
        }
    }
    __syncthreads();

    // coalesced outfeature stores: 32 consecutive px/row -> contiguous 256 B
    for (int i = tid; i < TILE * TILE; i += 256) {
        int r = i >> 5, c = i & 31;
        long gpix = (long)(Y + r) * WW + (X + c);
        *(float2*)(outf + gpix * 2) = *(float2*)&OUTS(i, 0);
    }
}

extern "C" void kernel_launch(void* const* d_in, const int* in_sizes, int n_in,
                              void* d_out, int out_size, void* d_ws, size_t ws_size,
                              hipStream_t stream) {
    const float* x   = (const float*)d_in[0];
    const float* x2  = (const float*)d_in[1];
    const float* w1a = (const float*)d_in[2];
    const float* b1a = (const float*)d_in[3];
    const float* w2a = (const float*)d_in[4];
    const float* b2a = (const float*)d_in[5];
    const float* w3a = (const float*)d_in[6];
    const float* b3a = (const float*)d_in[7];
    const float* w1b = (const float*)d_in[8];
    const float* b1b = (const float*)d_in[9];
    const float* w2b = (const float*)d_in[10];
    const float* b2b = (const float*)d_in[11];
    const float* w3b = (const float*)d_in[12];
    const float* b3b = (const float*)d_in[13];
    const float* lw1 = (const float*)d_in[14];
    const float* lb1 = (const float*)d_in[15];
    const float* lw2 = (const float*)d_in[16];
    const float* lb2 = (const float*)d_in[17];

    float* ws   = (float*)d_ws;
    float* outp = (float*)d_out;
    float* outfeature = outp;                         // [HW,2]
    float* den1 = outp + (size_t)2 * HH * WW;         // [H,W,6]
    float* den2 = outp + (size_t)8 * HH * WW;         // [H,W,6]

    prep_fold<<<1, 256, 0, stream>>>(lw1, lb1, lw2, lb2, ws);

    dim3 grid(WW / TILE, HH / TILE);
    encode_mlp<<<grid, 256, 0, stream>>>(x, x2,
                                         w1a, b1a, w2a, b2a, w3a, b3a,
                                         w1b, b1b, w2b, b2b, w3b, b3b,
                                         ws, outfeature, den1, den2);
}